// LinearTemporal_82832739270906
// MI455X (gfx1250) — compile-verified
//
#include <hip/hip_runtime.h>

typedef float v2f __attribute__((ext_vector_type(2)));
typedef float v8f __attribute__((ext_vector_type(8)));

#define PAST   24
#define FUTURE 24
#define B_DIM  32
#define L_DIM  512
#define D_DIM  256
#define NWIN   (L_DIM - PAST)   // 488

// out[b,i,f,d] = sum_p inp[b,i+p,d] * W[f,p] + bias[f]
// WMMA mapping: M=f (2 tiles of 16, second half-valid), N=d (16-chunks), K=p (6 x K=4 fp32 WMMA)
__global__ __launch_bounds__(256)
void LinearTemporal_wmma_f32(const float* __restrict__ inp,
                             const float* __restrict__ W,
                             const float* __restrict__ bias,
                             float* __restrict__ out)
{
    const int i    = blockIdx.x;          // window index 0..487
    const int b    = blockIdx.y;          // batch 0..31
    const int wave = threadIdx.x >> 5;    // 0..7
    const int lane = threadIdx.x & 31;
    const int lane16 = lane & 15;
    const int hi     = lane >> 4;         // 0 | 1 (half-wave)
    const int f_tile = wave & 1;          // which 16-row block of f
    const int dq     = wave >> 1;         // 0..3: quarter of the d dimension
    const int f_base = f_tile * 16;

    // ---- A operand: W tile (16f x 24p) as six 16x4 fp32 chunks ----
    // ISA layout 16x4 f32 A: lanes 0-15 -> {K=k0, k0+1}, lanes 16-31 -> {K=k0+2, k0+3}, M=lane%16
    const int f_row = (f_base + lane16 > FUTURE - 1) ? (FUTURE - 1) : (f_base + lane16);
    v2f a[6];
#pragma unroll
    for (int kc = 0; kc < 6; ++kc) {
        const int k0 = 4 * kc + 2 * hi;
        a[kc].x = W[f_row * PAST + k0];
        a[kc].y = W[f_row * PAST + k0 + 1];
    }

    // ---- C init = bias (bias add comes free through the accumulator) ----
    // D layout: VGPR r, lanes 0-15 -> M=r ; lanes 16-31 -> M=r+8 ; N=lane%16
    v8f cinit;
#pragma unroll
    for (int r = 0; r < 8; ++r) {
        const int f_c = (f_base + r + 8 * hi > FUTURE - 1) ? (FUTURE - 1)
                                                           : (f_base + r + 8 * hi);
        cinit[r] = bias[f_c];
    }

    // Single per-lane base addresses; all loop offsets become compile-time immediates
    // so the VMEM stream is base + ioffset (no per-load 64-bit VALU address math).
    const float* __restrict__ bbase =
        inp + (size_t)b * (L_DIM * D_DIM)
            + (size_t)(i + 2 * hi) * D_DIM + dq * 64 + lane16;

    float* __restrict__ sbase =
        out + ((size_t)b * NWIN + i) * (size_t)(FUTURE * D_DIM)
            + (size_t)(f_base + 8 * hi) * D_DIM + dq * 64 + lane16;

    // One uniform predicate for ALL stores of this lane:
    // tile 0: every lane's 8 rows are valid (f = r + 8*hi in 0..15)
    // tile 1: only lo half-wave rows are valid (f = 16 + r), hi half is padding
    const bool do_store = (f_tile == 0) | (hi == 0);

#pragma unroll
    for (int j = 0; j < 4; ++j) {
        v8f c = cinit;
#pragma unroll
        for (int kc = 0; kc < 6; ++kc) {
            // B operand 4x16 fp32: lanes 0-15 -> rows {k0,k0+1}, lanes 16-31 -> rows {k0+2,k0+3}
            v2f bm;
            bm.x = bbase[kc * 4 * D_DIM + j * 16];
            bm.y = bbase[kc * 4 * D_DIM + D_DIM + j * 16];
            // (neg_a, A, neg_b, B, c_mod, C, reuse_a, reuse_b)
            c = __builtin_amdgcn_wmma_f32_16x16x4_f32(
                    false, a[kc], false, bm, (short)0, c, false, false);
        }
        // ---- Stream the D tile out with non-temporal stores (keep L2 for the input).
        // Single exec-mask change guards all 8 stores.
        if (do_store) {
#pragma unroll
            for (int r = 0; r < 8; ++r) {
                __builtin_nontemporal_store(c[r], &sbase[r * D_DIM + j * 16]);
            }
        }
    }
}

extern "C" void kernel_launch(void* const* d_in, const int* in_sizes, int n_in,
                              void* d_out, int out_size, void* d_ws, size_t ws_size,
                              hipStream_t stream) {
    const float* inp  = (const float*)d_in[0];  // (32, 512, 256) fp32
    const float* W    = (const float*)d_in[1];  // (24, 24) fp32
    const float* bias = (const float*)d_in[2];  // (24,) fp32
    float* out = (float*)d_out;                 // (32, 488, 24, 256) fp32

    dim3 grid(NWIN, B_DIM);   // 488 x 32 blocks, 8 wave32s each
    LinearTemporal_wmma_f32<<<grid, 256, 0, stream>>>(inp, W, bias, out);
}